// FastNCC_730144441170
// MI455X (gfx1250) — compile-verified
//
#include <hip/hip_runtime.h>

// ---------------------------------------------------------------------------
// FastNCC for MI455X (gfx1250): separable 9^3 ones box filter, fused cc math,
// deterministic hierarchical reduction using layout-agnostic f32 WMMA.
// Problem dims (fixed by reference): B=2, C=1, D=160, H=192, W=160.
// ---------------------------------------------------------------------------

typedef __attribute__((ext_vector_type(2))) float v2f;
typedef __attribute__((ext_vector_type(8))) float v8f;

namespace ncc {
constexpr int cB = 2, cD = 160, cH = 192, cW = 160;
constexpr int cHW  = cH * cW;            // 30720
constexpr int cDHW = cD * cHW;           // 4,915,200
constexpr size_t cN = (size_t)cB * cDHW; // 9,830,400 voxels per map
constexpr int CH   = 40;                 // D-chunk per pass-2 block
constexpr int NBX  = cB * cH;            // 384
constexpr int NBY  = cD / CH;            // 4
constexpr int NPART = NBX * NBY;         // 1536 = 256 * 6
}

// Layout-agnostic wave32 WMMA reduction helper.
// Each lane contributes x via the B operand; A is all-ones. Every element of B
// feeds exactly 16 elements of D = A*B, independent of the (k,n) lane layout,
// so:  sum over all 32 lanes of the returned value == 16 * (wave sum of x).
__device__ __forceinline__ float wave_wmma_sum16(float x) {
  v2f a; a.x = 1.0f; a.y = 1.0f;   // A (16x4) = all ones in every lane slot
  v2f b; b.x = x;    b.y = 0.0f;   // B (4x16) holds each lane's partial once
  v8f c = {};
  c = __builtin_amdgcn_wmma_f32_16x16x4_f32(
      /*neg_a=*/false, a, /*neg_b=*/false, b,
      /*c_mod=*/(short)0, c, /*reuse_a=*/false, /*reuse_b=*/false);
  return c[0] + c[1] + c[2] + c[3] + c[4] + c[5] + c[6] + c[7];
}

// ---------------------------------------------------------------------------
// Pass 1: per (b,d) slice, 32x32 output tiles. Box-sum the 5 product maps
// along W then H (zero padding), write [5][B][D][H][W] workspace maps.
// ---------------------------------------------------------------------------
__global__ __launch_bounds__(256) void ncc_pass1(const float* __restrict__ I,
                                                 const float* __restrict__ J,
                                                 float* __restrict__ M) {
  using namespace ncc;
  const int w0 = blockIdx.x * 32;
  const int h0 = blockIdx.y * 32;
  const int bd = blockIdx.z;
  const int b  = bd / cD;
  const int d  = bd % cD;
  const int tid = threadIdx.x;

  __shared__ float tI[40 * 40];
  __shared__ float tJ[40 * 40];
  __shared__ float wsum[5][40 * 32];

  const float* Ib = I + (size_t)b * cDHW + (size_t)d * cHW;
  const float* Jb = J + (size_t)b * cDHW + (size_t)d * cHW;

  // Load 40x40 halo tile (zero padded at volume borders).
  for (int i = tid; i < 1600; i += 256) {
    const int hh = i / 40, ww = i - hh * 40;
    const int gh = h0 - 4 + hh, gw = w0 - 4 + ww;
    float vi = 0.0f, vj = 0.0f;
    if (gh >= 0 && gh < cH && gw >= 0 && gw < cW) {
      const int off = gh * cW + gw;
      vi = Ib[off];
      vj = Jb[off];
    }
    tI[i] = vi;
    tJ[i] = vj;
  }
  __syncthreads();

  // W-direction 9-tap sums of the 5 product maps: 40 rows x 32 cols.
  for (int i = tid; i < 40 * 32; i += 256) {
    const int hh = i >> 5, ww = i & 31;
    float s1 = 0.f, s2 = 0.f, s3 = 0.f, s4 = 0.f, s5 = 0.f;
#pragma unroll
    for (int k = 0; k < 9; ++k) {
      const float fa = tI[hh * 40 + ww + k];
      const float fb = tJ[hh * 40 + ww + k];
      s1 += fa; s2 += fb; s3 += fa * fa; s4 += fb * fb; s5 += fa * fb;
    }
    wsum[0][i] = s1; wsum[1][i] = s2; wsum[2][i] = s3;
    wsum[3][i] = s4; wsum[4][i] = s5;
  }
  __syncthreads();

  // H-direction 9-tap sums, write 5 maps.
  for (int i = tid; i < 32 * 32; i += 256) {
    const int hh = i >> 5, ww = i & 31;
    const size_t off = (size_t)b * cDHW + (size_t)d * cHW +
                       (size_t)(h0 + hh) * cW + (w0 + ww);
#pragma unroll
    for (int c5 = 0; c5 < 5; ++c5) {
      float s = 0.f;
#pragma unroll
      for (int k = 0; k < 9; ++k) s += wsum[c5][(hh + k) * 32 + ww];
      M[(size_t)c5 * cN + off] = s;
    }
  }
}

// ---------------------------------------------------------------------------
// Pass 2: D-direction running 9-tap sum (ring buffer in LDS), fused cc math,
// per-block partial via wave32 WMMA reduction + deterministic LDS tree.
// Block = 160 threads (5 full waves), one thread per W column.
// Loop peeled into {first iter | branch-free main | load-free tail} so the 5
// map loads issue unconditionally (clause-able, single wait) each iteration.
// ---------------------------------------------------------------------------
__global__ __launch_bounds__(160) void ncc_pass2(const float* __restrict__ M,
                                                 float* __restrict__ partials) {
  using namespace ncc;
  const int b  = blockIdx.x / cH;
  const int h  = blockIdx.x % cH;
  const int d0 = blockIdx.y * CH;
  const int w  = threadIdx.x;

  __shared__ float ring[5][9][160];  // per-thread 9-deep windows (own column only)
  __shared__ float red[160];

  const float* base0 = M + (size_t)b * cDHW + (size_t)h * cW + w;

  float S[5] = {0.f, 0.f, 0.f, 0.f, 0.f};

  // ---- Prologue: window sum over [d0-4, d0+3] (zero pad below 0). Uniform
  // once-per-block branch; both variants use unconditional loads.
  if (d0 == 0) {
#pragma unroll
    for (int dd = -4; dd < 0; ++dd) {
      const int slot = dd + 9;  // 5..8
#pragma unroll
      for (int c5 = 0; c5 < 5; ++c5) ring[c5][slot][w] = 0.f;
    }
    for (int dd = 0; dd < 4; ++dd) {
      const int slot = dd;  // 0..3
#pragma unroll
      for (int c5 = 0; c5 < 5; ++c5) {
        const float v = base0[(size_t)c5 * cN + (size_t)dd * cHW];
        ring[c5][slot][w] = v;
        S[c5] += v;
      }
    }
  } else {
    for (int dd = d0 - 4; dd <= d0 + 3; ++dd) {
      const int slot = dd % 9;
#pragma unroll
      for (int c5 = 0; c5 < 5; ++c5) {
        const float v = base0[(size_t)c5 * cN + (size_t)dd * cHW];
        ring[c5][slot][w] = v;
        S[c5] += v;
      }
    }
  }

  const float r9 = 1.0f / 9.0f;
  float acc = 0.f;
  int slot = (d0 + 4) % 9;

  // ---- First iteration (d = d0): load x[d0+4] (always < cD), no subtract.
  {
#pragma unroll
    for (int c5 = 0; c5 < 5; ++c5) {
      const float v = base0[(size_t)c5 * cN + (size_t)(d0 + 4) * cHW];
      ring[c5][slot][w] = v;
      S[c5] += v;
    }
    const float Is = S[0], Js = S[1], I2 = S[2], J2 = S[3], IJ = S[4];
    const float cross = IJ - Js * r9 * Is;
    const float Iv    = I2 - Is * r9 * Is;
    const float Jv    = J2 - Js * r9 * Js;
    acc += cross * cross / (Iv * Jv + 1e-5f);
    slot = (slot == 8) ? 0 : slot + 1;
  }

  // ---- Branch-free main body: d in (d0, dend), loads always in range.
  const int dend = (d0 + CH < cD - 4) ? (d0 + CH) : (cD - 4);  // uniform
  for (int d = d0 + 1; d < dend; ++d) {
    // Clamped prefetch ~8 rows ahead (select, not branch).
    const int dp = (d + 12 < cD) ? (d + 12) : (cD - 1);
#pragma unroll
    for (int c5 = 0; c5 < 5; ++c5)
      __builtin_prefetch(&base0[(size_t)c5 * cN + (size_t)dp * cHW], 0, 0);
#pragma unroll
    for (int c5 = 0; c5 < 5; ++c5) {
      const float old = ring[c5][slot][w];
      const float v = base0[(size_t)c5 * cN + (size_t)(d + 4) * cHW];
      ring[c5][slot][w] = v;
      S[c5] += v - old;
    }
    const float Is = S[0], Js = S[1], I2 = S[2], J2 = S[3], IJ = S[4];
    const float cross = IJ - Js * r9 * Is;
    const float Iv    = I2 - Is * r9 * Is;
    const float Jv    = J2 - Js * r9 * Js;
    acc += cross * cross / (Iv * Jv + 1e-5f);
    slot = (slot == 8) ? 0 : slot + 1;
  }

  // ---- Tail (only last chunk, d+4 >= cD): incoming values are zero pad.
  for (int d = dend; d < d0 + CH; ++d) {
#pragma unroll
    for (int c5 = 0; c5 < 5; ++c5) {
      S[c5] -= ring[c5][slot][w];  // ring still holds x[d-5]
    }
    const float Is = S[0], Js = S[1], I2 = S[2], J2 = S[3], IJ = S[4];
    const float cross = IJ - Js * r9 * Is;
    const float Iv    = I2 - Is * r9 * Is;
    const float Jv    = J2 - Js * r9 * Js;
    acc += cross * cross / (Iv * Jv + 1e-5f);
    slot = (slot == 8) ? 0 : slot + 1;
  }

  // WMMA in straight-line code: all 5 waves full EXEC. sum(p) = 16 * sum(acc).
  const float p = wave_wmma_sum16(acc);
  red[w] = p;
  __syncthreads();
  if (w < 80) red[w] += red[w + 80];
  __syncthreads();
  if (w < 40) red[w] += red[w + 40];
  __syncthreads();
  if (w < 20) red[w] += red[w + 20];
  __syncthreads();
  if (w < 10) red[w] += red[w + 10];
  __syncthreads();
  if (w < 5)  red[w] += red[w + 5];
  __syncthreads();
  if (w == 0) {
    const float t = red[0] + red[1] + red[2] + red[3] + red[4];
    partials[blockIdx.y * gridDim.x + blockIdx.x] = t * (1.0f / 16.0f);
  }
}

// ---------------------------------------------------------------------------
// Pass 3: reduce 1536 partials (fixed order -> deterministic), negate mean.
// ---------------------------------------------------------------------------
__global__ __launch_bounds__(256) void ncc_finish(const float* __restrict__ partials,
                                                  float* __restrict__ out) {
  using namespace ncc;
  __shared__ float red[256];
  const int tid = threadIdx.x;
  float s = 0.f;
#pragma unroll
  for (int k = 0; k < NPART / 256; ++k) s += partials[tid + 256 * k];

  const float p = wave_wmma_sum16(s);  // 8 full waves, straight-line
  red[tid] = p;
  __syncthreads();
  for (int off = 128; off >= 1; off >>= 1) {
    if (tid < off) red[tid] += red[tid + off];
    __syncthreads();
  }
  if (tid == 0) {
    const float total = red[0] * (1.0f / 16.0f);   // undo WMMA 16x weighting
    out[0] = -total * (1.0f / (float)cN);          // -mean(cc)
  }
}

// ---------------------------------------------------------------------------
extern "C" void kernel_launch(void* const* d_in, const int* in_sizes, int n_in,
                              void* d_out, int out_size, void* d_ws, size_t ws_size,
                              hipStream_t stream) {
  using namespace ncc;
  const float* I = (const float*)d_in[0];  // y_true
  const float* J = (const float*)d_in[1];  // y_pred
  float* M = (float*)d_ws;                 // 5 maps: 5 * 9,830,400 floats
  float* partials = M + 5 * cN;            // 1536 floats

  dim3 g1(cW / 32, cH / 32, cB * cD);      // (5, 6, 320)
  ncc_pass1<<<g1, dim3(256), 0, stream>>>(I, J, M);

  dim3 g2(NBX, NBY);                       // (384, 4)
  ncc_pass2<<<g2, dim3(160), 0, stream>>>(M, partials);

  ncc_finish<<<1, dim3(256), 0, stream>>>(partials, (float*)d_out);
}